// TNNColumnLayer_80771154969205
// MI455X (gfx1250) — compile-verified
//
#include <hip/hip_runtime.h>
#include <cmath>

#define ROWSN  29
#define COLSN  29
#define QN     16
#define RFN    16
#define NPREVN 4
#define PN     1024
#define HWN    128
#define THETA  200.0f

typedef float v2f __attribute__((ext_vector_type(2)));
typedef float v8f __attribute__((ext_vector_type(8)));
typedef float f4  __attribute__((ext_vector_type(4)));

__global__ __launch_bounds__(256) void tnn_column_kernel(
    const float* __restrict__ data,
    const float* __restrict__ weights,
    float* __restrict__ out_nl,     // (ROWS,COLS,Q)
    float* __restrict__ out_ist,    // (NUM,P)
    float* __restrict__ out_stdp)   // (NUM,P)
{
  __shared__ __align__(16) float t_nat[PN];   // window times, p = k*256 + i*16 + j
  __shared__ float t_srt[PN];                 // sorted times
  __shared__ int   idx_s[PN];                 // sort permutation
  __shared__ float chunkT[QN * 64];           // per-neuron 16-chunk sums (sorted order)
  __shared__ float ec_sh[QN], ec_out[QN], pot_out[QN], li_sh[QN];

  const int tid  = threadIdx.x;
  const int lane = tid & 31;
  const int wid  = tid >> 5;
  const int ncol = blockIdx.x;               // 0..840
  const int r0   = (ncol / COLSN) * 4;
  const int c0   = (ncol % COLSN) * 4;
  const size_t nbase = (size_t)ncol * QN;

  // Prefetch this column's 64KB weight block into L2 (global_prefetch_b8).
  {
    const char* wb = (const char*)(weights + nbase * PN);
    __builtin_prefetch(wb + (size_t)tid * 256, 0, 1);
  }

  // ---- Phase 1: gather window times into LDS (vectorized over NPREV) ----
  {
    const int i = tid >> 4, j = tid & 15;
    const f4 v = *(const f4*)(data + ((size_t)(r0 + i) * HWN + (c0 + j)) * NPREVN);
#pragma unroll
    for (int k = 0; k < 4; ++k) t_nat[k * 256 + tid] = v[k];
  }
  __syncthreads();
#pragma unroll
  for (int m = 0; m < 4; ++m) {
    const int p = tid + m * 256;
    t_srt[p] = t_nat[p];
    idx_s[p] = p;
  }
  __syncthreads();

  // ---- Phase 2: bitonic sort (key=time asc, payload=index); inf sorts last ----
  for (int k = 2; k <= PN; k <<= 1) {
    for (int j = k >> 1; j > 0; j >>= 1) {
#pragma unroll
      for (int m = 0; m < 4; ++m) {
        const int i   = tid + m * 256;
        const int ixj = i ^ j;
        if (ixj > i) {
          const bool up = ((i & k) == 0);
          const float a = t_srt[i], b = t_srt[ixj];
          if (up ? (a > b) : (a < b)) {
            t_srt[i] = b; t_srt[ixj] = a;
            const int ia = idx_s[i]; idx_s[i] = idx_s[ixj]; idx_s[ixj] = ia;
          }
        }
      }
      __syncthreads();
    }
  }

  // ---- Phase 3: 16-neuron x 64-chunk sums of sorted eff-weights via WMMA ----
  // T(16x64) = Wsorted(16x1024) * Indicator(1024x64), chunk = 16 sorted elems.
  // Tile per wave: wave w covers chunks [16w,16w+16), K in [256w, 256w+256).
  // Branch-free, software-pipelined A gather: loads are unconditional (indices
  // always valid); the inf-time masking is a VALU select on the loaded value,
  // and step s+1's loads overlap step s's WMMA.
  if (wid < 4) {
    const int cbase = wid * 16;
    const int kbase = wid * 256;
    const int q     = lane & 15;          // A-matrix M = lane mod 16 (both halves)
    const int koff  = (lane >> 4) * 2;    // lanes 16-31 hold K+2,K+3
    const float* wrow = weights + (nbase + q) * PN;

    v8f acc = {0.f,0.f,0.f,0.f,0.f,0.f,0.f,0.f};

    // preload step 0
    int   ka = kbase + koff;
    float t0 = t_srt[ka], t1 = t_srt[ka + 1];
    int   i0 = idx_s[ka], i1 = idx_s[ka + 1];
    float w0 = wrow[i0],  w1 = wrow[i1];

#pragma unroll 4
    for (int s = 0; s < 64; ++s) {
      v2f afrag;
      afrag[0] = (t0 < 1e30f) ? w0 : 0.0f;   // eff_w: 0 at inf times (cndmask)
      afrag[1] = (t1 < 1e30f) ? w1 : 0.0f;

      const int   cs = s >> 2;               // all 4 K's of this step share one chunk
      const float bv = ((lane & 15) == cs) ? 1.0f : 0.0f;  // identical-row B: layout-proof
      v2f bfrag; bfrag[0] = bv; bfrag[1] = bv;

      if (s < 63) {                          // prefetch next step's A data
        ka = kbase + (s + 1) * 4 + koff;
        t0 = t_srt[ka]; t1 = t_srt[ka + 1];
        i0 = idx_s[ka]; i1 = idx_s[ka + 1];
        w0 = wrow[i0];  w1 = wrow[i1];
      }

      acc = __builtin_amdgcn_wmma_f32_16x16x4_f32(
          false, afrag, false, bfrag, (short)0, acc, false, false);
    }
    // D layout: VGPR v -> M = v + 8*(lane>=16), N = lane&15
    const int hi = lane >> 4, n = lane & 15;
#pragma unroll
    for (int v = 0; v < 8; ++v)
      chunkT[(v + 8 * hi) * 64 + cbase + n] = acc[v];
  }
  __syncthreads();

  // ---- Phase 4: find first sorted position with cum >= THETA -> ec ----
  if (tid < QN) {
    const int q = tid;
    const float* wrow = weights + (nbase + q) * PN;
    float base = 0.0f;
    int cst = -1;
    for (int c = 0; c < 64; ++c) {
      const float nb = base + chunkT[q * 64 + c];
      if (nb >= THETA) { cst = c; break; }
      base = nb;
    }
    float ec = __builtin_inff();
    if (cst >= 0) {
      for (int k = cst * 16; k < PN; ++k) {
        const float tk = t_srt[k];
        const float wv = wrow[idx_s[k]];           // unconditional load
        base += (tk < 1e30f) ? wv : 0.0f;          // select, no branch
        if (base >= THETA) { ec = tk; break; }
      }
    }
    ec_sh[q] = ec;
  }
  __syncthreads();

  // ---- Phase 5: RNL refinement; each wave handles neurons wid and wid+8 ----
#pragma unroll
  for (int rep = 0; rep < 2; ++rep) {
    const int q = wid + rep * 8;
    const float ec = ec_sh[q];
    const float* wrow = weights + (nbase + q) * PN;
    float acc[7] = {0.f, 0.f, 0.f, 0.f, 0.f, 0.f, 0.f};
    for (int m = 0; m < 32; ++m) {
      const int p = lane + m * 32;          // natural order: coalesced weight reads
      const float tf = t_nat[p];
      const float w  = wrow[p];
      const float tw = (tf < 1e30f) ? rintf(w) : 0.0f;   // round-half-even, eff_w
#pragma unroll
      for (int t = 0; t < 7; ++t) {
        const float sim = ec + (float)t;
        const float r = (sim >= tf + tw) ? tw : ((sim < tf) ? 0.0f : (sim - tf + 1.0f));
        acc[t] += r;
      }
    }
#pragma unroll
    for (int t = 0; t < 7; ++t)
      for (int off = 16; off > 0; off >>= 1)
        acc[t] += __shfl_xor(acc[t], off, 32);
    if (lane == 0) {
      int st = -1;
#pragma unroll
      for (int t = 0; t < 7; ++t)
        if (st < 0 && acc[t] >= THETA) st = t;
      ec_out[q]  = (st >= 0) ? (float)st : __builtin_inff();
      pot_out[q] = (st >= 0) ? acc[st] : acc[0];
    }
  }
  __syncthreads();

  // ---- Phase 6: per-column WTA over Q ----
  if (tid == 0) {
    float mn = __builtin_inff();
    for (int q = 0; q < QN; ++q) mn = fminf(mn, ec_out[q]);
    float best = -1.0f; int mi = 0;
    for (int q = 0; q < QN; ++q) {
      const float m = (ec_out[q] == mn) ? pot_out[q] : 0.0f;
      if (m > best) { best = m; mi = q; }   // strict > keeps first max (argmax semantics)
    }
    for (int q = 0; q < QN; ++q)
      li_sh[q] = (q == mi) ? ec_out[mi] : __builtin_inff();
  }
  __syncthreads();

  // ---- Phase 7: outputs ----
  if (tid < QN) out_nl[ncol * QN + tid] = li_sh[tid];

  const f4 tv = ((const f4*)t_nat)[tid];   // 16B-aligned LDS read
#pragma unroll 4
  for (int q = 0; q < QN; ++q) {
    const size_t row = (nbase + q) * PN;
    const float lv = li_sh[q];
    f4 lv4; lv4[0] = lv; lv4[1] = lv; lv4[2] = lv; lv4[3] = lv;
    __builtin_nontemporal_store(tv,  (f4*)(out_ist  + row) + tid);
    __builtin_nontemporal_store(lv4, (f4*)(out_stdp + row) + tid);
  }
}

extern "C" void kernel_launch(void* const* d_in, const int* in_sizes, int n_in,
                              void* d_out, int out_size, void* d_ws, size_t ws_size,
                              hipStream_t stream) {
  const float* data    = (const float*)d_in[0];   // (128,128,4) f32
  const float* weights = (const float*)d_in[1];   // (NUM, P)   f32
  float* out = (float*)d_out;

  const size_t NUMv = (size_t)ROWSN * COLSN * QN;     // 13456
  float* out_nl   = out;                              // 13456
  float* out_ist  = out + NUMv;                       // NUM*P
  float* out_stdp = out + NUMv + NUMv * (size_t)PN;   // NUM*P

  dim3 grid(ROWSN * COLSN), block(256);
  hipLaunchKernelGGL(tnn_column_kernel, grid, block, 0, stream,
                     data, weights, out_nl, out_ist, out_stdp);
}